// TopKPool_30270929502677
// MI455X (gfx1250) — compile-verified
//
#include <hip/hip_runtime.h>

typedef float v2f __attribute__((ext_vector_type(2)));
typedef float v8f __attribute__((ext_vector_type(8)));

#define N_NODES 16384
#define F_DIM   256
#define K_TOP   8192   // ceil(0.5 * N)

// ---------------------------------------------------------------------------
// Kernel 1: scores = X @ score_vector via V_WMMA_F32_16X16X4_F32.
// 8 waves per block; each wave computes 16 rows' scores (64 WMMA steps, K=256).
//
// A fragment (16x4 f32, 2 VGPRs/lane, ISA 7.12.2):
//   lanes 0-15 : (X[row0+m][k0+0], X[row0+m][k0+1])
//   lanes 16-31: (X[row0+m][k0+2], X[row0+m][k0+3])
// B fragment (4x16 f32): we only consume column N=0 of D, and WMMA columns
// are independent, so columns 1..15 of B may hold garbage -> branchless:
//   every lane loads (s[k0+half], s[k0+2+half]); this puts the correct
//   (K,N=0) values in lanes 0 and 16 and harmless duplicates elsewhere.
// D (16x16 f32): column N=0 -> lane 0 holds M=0..7 in acc[0..7],
//   lane 16 holds M=8..15 in acc[0..7].
// ---------------------------------------------------------------------------
__global__ __launch_bounds__(256)
void topk_scores_wmma(const float* __restrict__ X,
                      const float* __restrict__ s,
                      float* __restrict__ scores) {
    __shared__ float ss[F_DIM];          // 1 KB: score vector staged in LDS
    const int tid = threadIdx.x;
    ss[tid] = s[tid];                    // blockDim == F_DIM == 256
    __syncthreads();

    const int lane = tid & 31;
    const int wave = tid >> 5;
    const int half = lane >> 4;          // 0: K pair {0,1}, 1: K pair {2,3}
    const int m    = lane & 15;          // row within 16-row tile
    const int row0 = (blockIdx.x * 8 + wave) * 16;

    v8f acc = {};
    const float* arow = X + (size_t)(row0 + m) * F_DIM + 2 * half;

#pragma unroll 4
    for (int k0 = 0; k0 < F_DIM; k0 += 4) {
        v2f a;                            // contiguous pair -> global_load_b64
        a.x = arow[k0 + 0];
        a.y = arow[k0 + 1];
        v2f b;                            // branchless, from LDS
        b.x = ss[k0 + half];
        b.y = ss[k0 + 2 + half];
        // (neg_a, A, neg_b, B, c_mod, C, reuse_a, reuse_b)
        acc = __builtin_amdgcn_wmma_f32_16x16x4_f32(
            false, a, false, b, (short)0, acc, false, false);
    }

    if ((lane & 15) == 0) {               // lanes 0 and 16 hold column N=0
#pragma unroll
        for (int v = 0; v < 8; ++v)
            scores[row0 + half * 8 + v] = acc[v];
    }
}

// ---------------------------------------------------------------------------
// Kernel 2: exact stable top-k via ranking. All 16384 scores live in 64 KB
// LDS (CDNA5 WGP has 320 KB). rank_i = #{ j : s_j > s_i ||
// (s_j == s_i && j < i) } reproduces lax.top_k's descending stable order.
// rank < K  =>  idx[rank] = i.
// ---------------------------------------------------------------------------
__global__ __launch_bounds__(256)
void topk_rank_select(const float* __restrict__ scores,
                      int* __restrict__ idx_ws,
                      float* __restrict__ idx_out) {
    __shared__ __align__(16) float sh[N_NODES];   // 64 KB
    const int tid = threadIdx.x;

    for (int j = tid; j < N_NODES / 4; j += 256)
        ((float4*)sh)[j] = ((const float4*)scores)[j];
    __syncthreads();

    const int   i  = blockIdx.x * 256 + tid;
    const float si = sh[i];

    int rank = 0;
    const float4* sh4 = (const float4*)sh;
    for (int j4 = 0; j4 < N_NODES / 4; ++j4) {
        float4 v = sh4[j4];
        int j = j4 * 4;
        rank += (v.x > si) || (v.x == si && (j + 0) < i);
        rank += (v.y > si) || (v.y == si && (j + 1) < i);
        rank += (v.z > si) || (v.z == si && (j + 2) < i);
        rank += (v.w > si) || (v.w == si && (j + 3) < i);
    }
    if (rank < K_TOP) {
        idx_ws[rank]  = i;
        idx_out[rank] = (float)i;
    }
}

// ---------------------------------------------------------------------------
// Kernel 3: X_pooled[r] = X[idx[r]]  (float4-wide row copy, 1 block per row)
// ---------------------------------------------------------------------------
__global__ __launch_bounds__(64)
void topk_gather_x(const float* __restrict__ X,
                   const int* __restrict__ idx,
                   float* __restrict__ Xp) {
    const int r   = blockIdx.x;
    const int src = idx[r];
    const float4* in  = (const float4*)(X  + (size_t)src * F_DIM);
    float4*       out = (float4*)      (Xp + (size_t)r   * F_DIM);
    out[threadIdx.x] = in[threadIdx.x];
}

// ---------------------------------------------------------------------------
// Kernel 4: A_pooled[r][c] = A[idx[r]][idx[c]]. One block per output row;
// column index vector (32 KB) staged in LDS via int4 copies. This is the
// bandwidth-dominant kernel (~536 MB traffic -> ~23 us at 23.3 TB/s HBM).
// ---------------------------------------------------------------------------
__global__ __launch_bounds__(256)
void topk_gather_a(const float* __restrict__ A,
                   const int* __restrict__ idx,
                   float* __restrict__ Ap) {
    __shared__ __align__(16) int sidx[K_TOP];     // 32 KB
    const int tid = threadIdx.x;

    for (int j = tid; j < K_TOP / 4; j += 256)
        ((int4*)sidx)[j] = ((const int4*)idx)[j];
    __syncthreads();

    const size_t src_base = (size_t)sidx[blockIdx.x] * N_NODES;
    float* out = Ap + (size_t)blockIdx.x * K_TOP;

    for (int c = tid; c < K_TOP; c += 256)
        out[c] = A[src_base + (size_t)sidx[c]];
}

// ---------------------------------------------------------------------------
extern "C" void kernel_launch(void* const* d_in, const int* in_sizes, int n_in,
                              void* d_out, int out_size, void* d_ws, size_t ws_size,
                              hipStream_t stream) {
    const float* X = (const float*)d_in[0];           // [N, F]
    const float* A = (const float*)d_in[1];           // [N, N]
    const float* s = (const float*)d_in[2];           // [F, 1]

    float* out     = (float*)d_out;
    float* Xp      = out;                                        // [K, F]
    float* Ap      = out + (size_t)K_TOP * F_DIM;                // [K, K]
    float* idx_out = Ap  + (size_t)K_TOP * K_TOP;                // [K]

    float* scores = (float*)d_ws;                                // 64 KB
    int*   idx_ws = (int*)((char*)d_ws + (size_t)N_NODES * sizeof(float)); // 32 KB

    topk_scores_wmma<<<N_NODES / 128, 256, 0, stream>>>(X, s, scores);
    topk_rank_select<<<N_NODES / 256, 256, 0, stream>>>(scores, idx_ws, idx_out);
    topk_gather_x<<<K_TOP, 64, 0, stream>>>(X, idx_ws, Xp);
    topk_gather_a<<<K_TOP, 256, 0, stream>>>(A, idx_ws, Ap);
}